// EnergyConditionedAtomAttention_34797825032826
// MI455X (gfx1250) — compile-verified
//
#include <hip/hip_runtime.h>
#include <math.h>
#include <stdint.h>

// ---- problem constants (from reference) ----
#define BB   32
#define NN   256
#define NE_  256
#define AD_  128
#define ED_  32
#define HID_ 256
#define LAT_ 256
#define RBF_ 16
#define ZE_  32
#define NH_  8
#define HD_  32
#define E_ATT_ 4096
#define KIN_  177      // AD + ZE + 1 + RBF
#define KPAD_ 192      // padded to multiple of 32
#define QIN_  160      // AD + ED
#define ROWS_ (BB*NN)  // 8192
#define QROWS_ (BB*NE_) // 8192

typedef __attribute__((ext_vector_type(2))) float v2f;
typedef __attribute__((ext_vector_type(8))) float v8f;

// ---- CDNA5 async global->LDS copy (no VGPR round-trip, tracked by ASYNCcnt) ----
__device__ __forceinline__ void async_copy_b128(unsigned lds_byte_addr, const float* gaddr) {
    asm volatile("global_load_async_to_lds_b128 %0, %1, off"
                 :: "v"(lds_byte_addr),
                    "v"((unsigned long long)(uintptr_t)gaddr)
                 : "memory");
}

__device__ __forceinline__ void wait_asynccnt0() {
#if __has_builtin(__builtin_amdgcn_s_wait_asynccnt)
    __builtin_amdgcn_s_wait_asynccnt(0);
#else
    asm volatile("s_wait_asynccnt 0x0" ::: "memory");
#endif
}

__device__ __forceinline__ void wait_asynccnt4() {
#if __has_builtin(__builtin_amdgcn_s_wait_asynccnt)
    __builtin_amdgcn_s_wait_asynccnt(4);
#else
    asm volatile("s_wait_asynccnt 0x4" ::: "memory");
#endif
}

// ---------------- prep kernels ----------------

__global__ void zero_small_kernel(float* dbuf, float* attm, int n) {
    int i = blockIdx.x * blockDim.x + threadIdx.x;
    if (i < n) { dbuf[i] = 0.0f; attm[i] = 0.0f; }
}

__global__ void scatter_kernel(const int* __restrict__ att_dst,
                               const float* __restrict__ att_dist,
                               const unsigned char* __restrict__ mask,
                               float* dbuf, float* attm, int ne) {
    int i = blockIdx.x * blockDim.x + threadIdx.x;
    if (i < ne) {
        int t = att_dst[i];
        dbuf[t] = att_dist[i];
        attm[t] = mask[t] ? 1.0f : 0.0f;
    }
}

__global__ void build_qin_kernel(const float* __restrict__ h,
                                 const float* __restrict__ e_feat,
                                 const int* __restrict__ abs_idx,
                                 float* __restrict__ qin) {
    int idx = blockIdx.x * blockDim.x + threadIdx.x;
    int total = QROWS_ * QIN_;
    if (idx >= total) return;
    int row = idx / QIN_;
    int col = idx - row * QIN_;
    int b = row / NE_;
    int e = row - b * NE_;
    float v;
    if (col < AD_) {
        int an = abs_idx[b];
        v = h[((size_t)(b * NN + an)) * AD_ + col];
    } else {
        v = e_feat[(size_t)e * ED_ + (col - AD_)];
    }
    qin[(size_t)row * QIN_ + col] = v;
}

__global__ void build_atom_kernel(const float* __restrict__ h,
                                  const int* __restrict__ z,
                                  const int* __restrict__ abs_idx,
                                  const float* __restrict__ zemb,
                                  const float* __restrict__ dbuf,
                                  float* __restrict__ atom) {
    int idx = blockIdx.x * blockDim.x + threadIdx.x;
    int total = ROWS_ * KPAD_;
    if (idx >= total) return;
    int row = idx / KPAD_;
    int col = idx - row * KPAD_;
    int b = row / NN;
    int n = row - b * NN;
    float v = 0.0f;
    if (col < AD_) {
        v = h[(size_t)row * AD_ + col];
    } else if (col < AD_ + ZE_) {
        v = zemb[(size_t)z[row] * ZE_ + (col - AD_)];
    } else if (col == AD_ + ZE_) {
        v = (n == abs_idx[b]) ? 1.0f : 0.0f;
    } else if (col < KIN_) {
        int j = col - (AD_ + ZE_ + 1);
        float off = (float)j * (5.0f / 15.0f);
        float dx = dbuf[row] - off;
        v = __expf(-4.5f * dx * dx);   // coeff = -0.5/(CUT/15)^2
    }
    atom[(size_t)row * KPAD_ + col] = v;
}

__global__ void env_kernel(const float* __restrict__ dbuf, float* __restrict__ env, int n) {
    int i = blockIdx.x * blockDim.x + threadIdx.x;
    if (i < n) {
        float d = dbuf[i];
        float e = 0.5f * (__cosf(3.14159265358979323846f * d / 5.0f) + 1.0f);
        env[i] = (d < 5.0f) ? e : 0.0f;
    }
}

// Pad (rows Ksrc..Kdst-1 = 0) and pair-interleave weights:
// dst[(k/2)*(2*256) + c*2 + (k&1)] = src[k*256 + c]
__global__ void interleave_w_kernel(const float* __restrict__ src, float* __restrict__ dst,
                                    int Ksrc, int Kdst) {
    int idx = blockIdx.x * blockDim.x + threadIdx.x;
    int total = Kdst * HID_;
    if (idx >= total) return;
    int k = idx / HID_;
    int c = idx - k * HID_;
    float v = (k < Ksrc) ? src[(size_t)k * HID_ + c] : 0.0f;
    dst[(size_t)(k >> 1) * (2 * HID_) + c * 2 + (k & 1)] = v;
}

// ---------------- WMMA f32 GEMM, async-LDS staged, double-buffered -------
// C = act(A[MxK] @ Wi + bias) (* rowscale), Wi = pair-interleaved K x 256 weights.
// block = 256 threads (8 waves); WG tile 128(M) x 128(N); K chunked by 16,
// double-buffered: chunk c+1 DMA overlaps chunk c compute.
// Wave (w): M strip = (w&3)*32, N strip = (w>>2)*64 -> 2x4 wmma tiles.
// grid = (N/128, M/128). Requires: M%128==0, N==256 here, K%16==0, lda%4==0.

#define SA_  20   // A LDS row stride: 80 B rows (16B aligned), bank-conflict free
#define SBI_ 288  // B LDS pair-row stride: 1152 B (16B aligned), 288%64==32 -> halves disjoint

__global__ __launch_bounds__(256) void gemm_wmma_f32(
    const float* __restrict__ A, int lda,
    const float* __restrict__ Wi,         // (K/2) x (2N) pair-interleaved
    const float* __restrict__ bias,
    float* __restrict__ C,
    const float* __restrict__ rowscale,   // may be null
    int N, int K, int act) {
    __shared__ __align__(16) float As[2][128 * SA_];
    __shared__ __align__(16) float Bs[2][8 * SBI_];

    const int tid  = threadIdx.x;
    const int lane = tid & 31;
    const int wave = tid >> 5;
    const int half = lane >> 4;
    const int l16  = lane & 15;
    const int m_blk = blockIdx.y * 128;
    const int n_blk = blockIdx.x * 128;
    const int m_wave = (wave & 3) * 32;
    const int n_wave = (wave >> 2) * 64;

    v8f acc[2][4] = {};

    const int nc = K >> 4;   // number of 16-wide K chunks

    // issue DMA for chunk 0 into buffer 0 (4 async b128 per thread: 2 A + 2 B)
    {
        #pragma unroll
        for (int i = 0; i < 2; ++i) {
            int f = tid + 256 * i;            // A: 128 rows x 16 floats = 512 x b128
            int row = f >> 2;
            int kq  = (f & 3) << 2;
            async_copy_b128((unsigned)(uintptr_t)&As[0][row * SA_ + kq],
                            A + (size_t)(m_blk + row) * lda + kq);
        }
        #pragma unroll
        for (int i = 0; i < 2; ++i) {
            int f = tid + 256 * i;            // B: 8 pair-rows x 256 floats = 512 x b128
            int pr = f >> 6;
            int cc = (f & 63) << 2;
            async_copy_b128((unsigned)(uintptr_t)&Bs[0][pr * SBI_ + cc],
                            Wi + (size_t)pr * (2 * N) + n_blk * 2 + cc);
        }
    }

    for (int c = 0; c < nc; ++c) {
        const int cur = c & 1;
        const int nxt = cur ^ 1;
        if (c + 1 < nc) {
            // prefetch chunk c+1 into other buffer (overlaps with wait+compute)
            const int k0n = (c + 1) << 4;
            #pragma unroll
            for (int i = 0; i < 2; ++i) {
                int f = tid + 256 * i;
                int row = f >> 2;
                int kq  = (f & 3) << 2;
                async_copy_b128((unsigned)(uintptr_t)&As[nxt][row * SA_ + kq],
                                A + (size_t)(m_blk + row) * lda + k0n + kq);
            }
            #pragma unroll
            for (int i = 0; i < 2; ++i) {
                int f = tid + 256 * i;
                int pr = f >> 6;
                int cc = (f & 63) << 2;
                async_copy_b128((unsigned)(uintptr_t)&Bs[nxt][pr * SBI_ + cc],
                                Wi + (size_t)((k0n >> 1) + pr) * (2 * N) + n_blk * 2 + cc);
            }
            wait_asynccnt4();   // ASYNCcnt retires in order: chunk c's 4 ops are done
        } else {
            wait_asynccnt0();
        }
        __syncthreads();        // every wave's DMA for chunk c has landed

        // ---- compute 4 k-steps from LDS: pure ds_load_b64 -> wmma ----
        #pragma unroll
        for (int kk = 0; kk < 16; kk += 4) {
            const int ka = kk + 2 * half;             // ISA A/B frag layout
            v2f a0 = *(const v2f*)&As[cur][(m_wave +      l16) * SA_ + ka];
            v2f a1 = *(const v2f*)&As[cur][(m_wave + 16 + l16) * SA_ + ka];
            #pragma unroll
            for (int t = 0; t < 4; ++t) {
                int col = n_wave + t * 16 + l16;
                v2f bf = *(const v2f*)&Bs[cur][(ka >> 1) * SBI_ + col * 2]; // {W[ka][col], W[ka+1][col]}
                acc[0][t] = __builtin_amdgcn_wmma_f32_16x16x4_f32(false, a0, false, bf, (short)0, acc[0][t], false, false);
                acc[1][t] = __builtin_amdgcn_wmma_f32_16x16x4_f32(false, a1, false, bf, (short)0, acc[1][t], false, false);
            }
        }
        __syncthreads();        // all waves done reading buf[cur] before it is re-DMA'd
    }

    // ---- epilogue: bias + SiLU + optional row scale ----
    #pragma unroll
    for (int ms = 0; ms < 2; ++ms) {
        #pragma unroll
        for (int t = 0; t < 4; ++t) {
            int col = n_blk + n_wave + t * 16 + l16;
            float bv = bias[col];
            #pragma unroll
            for (int r = 0; r < 8; ++r) {
                int row = m_blk + m_wave + ms * 16 + r + 8 * half;  // C/D layout
                float x = acc[ms][t][r] + bv;
                if (act) x = x * (1.0f / (1.0f + __expf(-x)));       // SiLU
                if (rowscale) x *= rowscale[row];
                C[(size_t)row * N + col] = x;
            }
        }
    }
}

// ---------------- attention: one WG per (b,e); one wave per head ------------

__global__ __launch_bounds__(256) void attention_kernel(
    const float* __restrict__ q,    // (B,NE,256) rows, head h at cols [32h,32h+32)
    const float* __restrict__ k,    // (B,N,256)
    const float* __restrict__ v,    // (B,N,256), env pre-applied
    const float* __restrict__ attm, // (B,N) 0/1 floats
    float* __restrict__ out) {      // (B,NE,256)
    __shared__ float psh[NH_][NN];
    const int e = blockIdx.x;
    const int b = blockIdx.y;
    const int h = threadIdx.x >> 5;
    const int lane = threadIdx.x & 31;

    const float* qp = q + ((size_t)(b * NE_ + e)) * LAT_ + h * HD_;
    float qreg[HD_];
    #pragma unroll
    for (int d = 0; d < HD_; ++d) qreg[d] = qp[d];

    float sv[8];
    float smax = -3.0e38f;
    #pragma unroll
    for (int j = 0; j < 8; ++j) {
        int n = lane + 32 * j;
        const float* kp = k + ((size_t)(b * NN + n)) * LAT_ + h * HD_;
        float s = 0.0f;
        #pragma unroll
        for (int d = 0; d < HD_; ++d) s += qreg[d] * kp[d];
        s *= 0.17677669529663687f;  // 1/sqrt(32)
        float am = attm[b * NN + n];
        sv[j] = (am > 0.0f) ? s : -1.0e9f;
        smax = fmaxf(smax, sv[j]);
    }
    #pragma unroll
    for (int off = 16; off > 0; off >>= 1) smax = fmaxf(smax, __shfl_xor(smax, off, 32));

    float dsum = 0.0f;
    #pragma unroll
    for (int j = 0; j < 8; ++j) {
        int n = lane + 32 * j;
        float am = attm[b * NN + n];
        float p = (am > 0.0f) ? __expf(sv[j] - smax) : 0.0f;
        psh[h][n] = p;
        dsum += p;
    }
    #pragma unroll
    for (int off = 16; off > 0; off >>= 1) dsum += __shfl_xor(dsum, off, 32);
    __syncthreads();

    const float inv = 1.0f / fmaxf(dsum, 1e-8f);
    const float* vp = v + ((size_t)(b * NN)) * LAT_ + h * HD_ + lane;
    float acc = 0.0f;
    for (int n = 0; n < NN; ++n) acc += psh[h][n] * vp[(size_t)n * LAT_];
    out[((size_t)(b * NE_ + e)) * LAT_ + h * HD_ + lane] = acc * inv;
}

// ---------------- launch ----------------

static inline void run_gemm(const float* A, int lda, const float* Wi, const float* bias,
                            float* C, const float* rowscale, int M, int N, int K, int act,
                            hipStream_t stream) {
    dim3 grid(N / 128, M / 128);
    gemm_wmma_f32<<<grid, 256, 0, stream>>>(A, lda, Wi, bias, C, rowscale, N, K, act);
}

static inline void run_interleave(const float* src, float* dst, int Ksrc, int Kdst,
                                  hipStream_t stream) {
    int total = Kdst * HID_;
    interleave_w_kernel<<<(total + 255) / 256, 256, 0, stream>>>(src, dst, Ksrc, Kdst);
}

extern "C" void kernel_launch(void* const* d_in, const int* in_sizes, int n_in,
                              void* d_out, int out_size, void* d_ws, size_t ws_size,
                              hipStream_t stream) {
    const float* h        = (const float*)d_in[0];
    const int*   z        = (const int*)d_in[1];
    const unsigned char* mask = (const unsigned char*)d_in[2];
    const float* e_feat   = (const float*)d_in[3];
    const int*   abs_idx  = (const int*)d_in[4];
    const int*   att_dst  = (const int*)d_in[5];
    const float* att_dist = (const float*)d_in[6];
    const float* zemb     = (const float*)d_in[7];
    const float* qW1 = (const float*)d_in[8];  const float* qb1 = (const float*)d_in[9];
    const float* qW2 = (const float*)d_in[10]; const float* qb2 = (const float*)d_in[11];
    const float* qW3 = (const float*)d_in[12]; const float* qb3 = (const float*)d_in[13];
    const float* kW1 = (const float*)d_in[14]; const float* kb1 = (const float*)d_in[15];
    const float* kW2 = (const float*)d_in[16]; const float* kb2 = (const float*)d_in[17];
    const float* kW3 = (const float*)d_in[18]; const float* kb3 = (const float*)d_in[19];
    const float* vW1 = (const float*)d_in[20]; const float* vb1 = (const float*)d_in[21];
    const float* vW2 = (const float*)d_in[22]; const float* vb2 = (const float*)d_in[23];
    const float* vW3 = (const float*)d_in[24]; const float* vb3 = (const float*)d_in[25];
    const float* oW1 = (const float*)d_in[26]; const float* ob1 = (const float*)d_in[27];
    const float* oW2 = (const float*)d_in[28]; const float* ob2 = (const float*)d_in[29];
    float* outp = (float*)d_out;

    float* ws = (float*)d_ws;
    // workspace layout (floats)
    float* dbuf  = ws + 0;                  //  8192
    float* attm  = ws + 8192;               //  8192
    float* env   = ws + 16384;               //  8192
    float* wi    = ws + 24576;              // interleaved weights region (663552)
    float* qW1i  = wi + 0;                  // 160*256
    float* qW2i  = wi + 40960;              // 256*256
    float* qW3i  = wi + 106496;
    float* kW1i  = wi + 172032;             // 192*256
    float* kW2i  = wi + 221184;
    float* kW3i  = wi + 286720;
    float* vW1i  = wi + 352256;             // 192*256
    float* vW2i  = wi + 401408;
    float* vW3i  = wi + 466944;
    float* oW1i  = wi + 532480;
    float* oW2i  = wi + 598016;
    float* qin   = ws + 688128;             // 8192*160
    float* atom  = ws + 1998848;            // 8192*192
    float* t1    = ws + 3571712;            // 8192*256
    float* t2    = ws + 5668864;            // 8192*256
    float* qb    = ws + 7766016;            // 8192*256
    float* kb    = ws + 9863168;            // 8192*256
    float* vb    = ws + 11960320;           // 8192*256
    (void)ws_size; (void)n_in; (void)in_sizes; (void)out_size;

    // 1) d / att_mask
    zero_small_kernel<<<(ROWS_ + 255) / 256, 256, 0, stream>>>(dbuf, attm, ROWS_);
    scatter_kernel<<<(E_ATT_ + 255) / 256, 256, 0, stream>>>(att_dst, att_dist, mask, dbuf, attm, E_ATT_);

    // 2) feature assembly
    {
        int total = QROWS_ * QIN_;
        build_qin_kernel<<<(total + 255) / 256, 256, 0, stream>>>(h, e_feat, abs_idx, qin);
    }
    {
        int total = ROWS_ * KPAD_;
        build_atom_kernel<<<(total + 255) / 256, 256, 0, stream>>>(h, z, abs_idx, zemb, dbuf, atom);
    }
    env_kernel<<<(ROWS_ + 255) / 256, 256, 0, stream>>>(dbuf, env, ROWS_);

    // 2b) pad + pair-interleave all weight matrices (tiny, one-time per call)
    run_interleave(qW1, qW1i, QIN_,  QIN_,  stream);
    run_interleave(qW2, qW2i, HID_,  HID_,  stream);
    run_interleave(qW3, qW3i, HID_,  HID_,  stream);
    run_interleave(kW1, kW1i, KIN_,  KPAD_, stream);
    run_interleave(kW2, kW2i, HID_,  HID_,  stream);
    run_interleave(kW3, kW3i, HID_,  HID_,  stream);
    run_interleave(vW1, vW1i, KIN_,  KPAD_, stream);
    run_interleave(vW2, vW2i, HID_,  HID_,  stream);
    run_interleave(vW3, vW3i, HID_,  HID_,  stream);
    run_interleave(oW1, oW1i, LAT_,  LAT_,  stream);
    run_interleave(oW2, oW2i, HID_,  HID_,  stream);

    // 3) MLPs (all WMMA f32, async-LDS staged, double-buffered)
    // K chain
    run_gemm(atom, KPAD_, kW1i, kb1, t1, nullptr, ROWS_, HID_, KPAD_, 1, stream);
    run_gemm(t1, HID_, kW2i, kb2, t2, nullptr, ROWS_, HID_, HID_, 1, stream);
    run_gemm(t2, HID_, kW3i, kb3, kb, nullptr, ROWS_, LAT_, HID_, 0, stream);
    // V chain (env fused into last layer as row scale)
    run_gemm(atom, KPAD_, vW1i, vb1, t1, nullptr, ROWS_, HID_, KPAD_, 1, stream);
    run_gemm(t1, HID_, vW2i, vb2, t2, nullptr, ROWS_, HID_, HID_, 1, stream);
    run_gemm(t2, HID_, vW3i, vb3, vb, env, ROWS_, LAT_, HID_, 0, stream);
    // Q chain
    run_gemm(qin, QIN_, qW1i, qb1, t1, nullptr, QROWS_, HID_, QIN_, 1, stream);
    run_gemm(t1, HID_, qW2i, qb2, t2, nullptr, QROWS_, HID_, HID_, 1, stream);
    run_gemm(t2, HID_, qW3i, qb3, qb, nullptr, QROWS_, LAT_, HID_, 0, stream);

    // 4) attention -> t1
    {
        dim3 grid(NE_, BB);
        attention_kernel<<<grid, 256, 0, stream>>>(qb, kb, vb, attm, t1);
    }

    // 5) output MLP -> d_out
    run_gemm(t1, LAT_, oW1i, ob1, t2, nullptr, QROWS_, HID_, LAT_, 1, stream);
    run_gemm(t2, HID_, oW2i, ob2, outp, nullptr, QROWS_, LAT_, HID_, 0, stream);
}